// TSPDLNodeModel_8830452760706
// MI455X (gfx1250) — compile-verified
//
#include <hip/hip_runtime.h>

#define N_NODES 50000
#define N_EDGES 640000
#define EMBED   128
#define ROW_TILES (N_NODES / 16)   // 3125, exact

typedef __attribute__((ext_vector_type(2))) float v2f;
typedef __attribute__((ext_vector_type(8))) float v8f;

// ---------------- utility kernels ----------------

__global__ __launch_bounds__(256) void k_zero(float* p, int n) {
    int i = blockIdx.x * blockDim.x + threadIdx.x;
    if (i < n) p[i] = 0.0f;
}

__global__ __launch_bounds__(256) void k_degrees(const int* __restrict__ src,
                                                 const int* __restrict__ dst,
                                                 float* __restrict__ outdeg,
                                                 float* __restrict__ indeg, int nE) {
    int e = blockIdx.x * blockDim.x + threadIdx.x;
    if (e < nE) {
        atomicAdd(&outdeg[src[e]], 1.0f);
        atomicAdd(&indeg[dst[e]], 1.0f);
    }
}

// in-place deg -> deg^-0.5 with clamp(>=1)
__global__ __launch_bounds__(256) void k_norm(float* __restrict__ outdeg,
                                              float* __restrict__ indeg, int n) {
    int i = blockIdx.x * blockDim.x + threadIdx.x;
    if (i < n) {
        outdeg[i] = rsqrtf(fmaxf(outdeg[i], 1.0f));
        indeg[i]  = rsqrtf(fmaxf(indeg[i], 1.0f));
    }
}

// ---------------- embed GEMM: h = nf[50000x4] @ W[4x128] + b ----------------
// One wave per 16-row tile; IN_DIM=4 == one V_WMMA_F32_16X16X4_F32 per 16-col tile.
__global__ __launch_bounds__(256) void k_embed(const float* __restrict__ nf,
                                               const float* __restrict__ W,
                                               const float* __restrict__ bias,
                                               float* __restrict__ h) {
    int wave = (blockIdx.x * blockDim.x + threadIdx.x) >> 5;
    int lane = threadIdx.x & 31;
    if (wave >= ROW_TILES) return;             // uniform per wave -> EXEC all-1s for WMMA
    int half = lane >> 4;                      // 0: K=0,1   1: K=2,3
    int lrow = lane & 15;
    int row  = wave * 16 + lrow;
    int k0   = 2 * half;

    v2f a;
    a.x = nf[row * 4 + k0];
    a.y = nf[row * 4 + k0 + 1];

    #pragma unroll
    for (int t = 0; t < 8; ++t) {
        int col = t * 16 + lrow;
        v2f b;
        b.x = W[(k0 + 0) * EMBED + col];
        b.y = W[(k0 + 1) * EMBED + col];
        float bv = bias[col];
        v8f c;
        #pragma unroll
        for (int r = 0; r < 8; ++r) c[r] = bv;
        c = __builtin_amdgcn_wmma_f32_16x16x4_f32(false, a, false, b, (short)0, c,
                                                  false, false);
        #pragma unroll
        for (int r = 0; r < 8; ++r)
            h[(wave * 16 + r + 8 * half) * EMBED + col] = c[r];
    }
}

// ---------------- SpMM scatter: agg[dst] += h[src] * norm_src[src] ----------------
// 32 lanes x float4 = one 128-wide edge row per wave; fully coalesced.
__global__ __launch_bounds__(256) void k_scatter(const float* __restrict__ h,
                                                 const float* __restrict__ norm_src,
                                                 const int* __restrict__ src,
                                                 const int* __restrict__ dst,
                                                 float* __restrict__ agg, int nE) {
    int idx = blockIdx.x * blockDim.x + threadIdx.x;
    int e = idx >> 5;
    int g = idx & 31;
    if (e >= nE) return;
    int s = src[e], d = dst[e];
    float ns = norm_src[s];
    float4 v = ((const float4*)(h + (size_t)s * EMBED))[g];
    float* ap = agg + (size_t)d * EMBED + g * 4;
    atomicAdd(ap + 0, v.x * ns);
    atomicAdd(ap + 1, v.y * ns);
    atomicAdd(ap + 2, v.z * ns);
    atomicAdd(ap + 3, v.w * ns);
}

// ---------------- 128x128 GEMM: out = act((in .* norm?) @ W + bias) ----------------
// 8 waves/block, wave w owns 16-col strip; B frags resident in 64 VGPRs;
// A tile (16x128) staged in LDS (fused norm_dst scaling); 32 f32 WMMAs / tile.
__global__ __launch_bounds__(256) void k_gemm128(const float* __restrict__ in,
                                                 const float* __restrict__ norm,
                                                 const float* __restrict__ W,
                                                 const float* __restrict__ bias,
                                                 float* __restrict__ out,
                                                 int nTiles, int relu) {
    __shared__ float sA[16 * EMBED];
    int lane = threadIdx.x & 31;
    int w    = threadIdx.x >> 5;
    int half = lane >> 4;
    int lrow = lane & 15;
    int col  = w * 16 + lrow;

    // Load this wave's B fragments for the whole kernel (W is [128 x 128] row-major).
    v2f bf[32];
    #pragma unroll
    for (int kc = 0; kc < 32; ++kc) {
        int k = kc * 4 + 2 * half;
        bf[kc].x = W[(k + 0) * EMBED + col];
        bf[kc].y = W[(k + 1) * EMBED + col];
    }
    float bv = bias[col];

    for (int rt = blockIdx.x; rt < nTiles; rt += gridDim.x) {
        // cooperative, coalesced A-tile load with fused source scaling
        #pragma unroll
        for (int j = 0; j < 8; ++j) {
            int el   = threadIdx.x + j * 256;
            int r    = el >> 7;
            int cc   = el & 127;
            int grow = rt * 16 + r;
            float v  = in[(size_t)grow * EMBED + cc];
            if (norm) v *= norm[grow];
            sA[el] = v;
        }
        __syncthreads();

        v8f c;
        #pragma unroll
        for (int r = 0; r < 8; ++r) c[r] = bv;
        #pragma unroll
        for (int kc = 0; kc < 32; ++kc) {
            int k = kc * 4 + 2 * half;
            v2f a;
            a.x = sA[lrow * EMBED + k];
            a.y = sA[lrow * EMBED + k + 1];
            c = __builtin_amdgcn_wmma_f32_16x16x4_f32(false, a, false, bf[kc],
                                                      (short)0, c, false, false);
        }
        #pragma unroll
        for (int r = 0; r < 8; ++r) {
            float v = c[r];
            if (relu) v = fmaxf(v, 0.0f);
            out[(size_t)(rt * 16 + r + 8 * half) * EMBED + col] = v;
        }
        __syncthreads();
    }
}

// ---------------- final projection: out[n] = h[n,:] . W_o2 + b ----------------
__global__ __launch_bounds__(256) void k_out2(const float* __restrict__ h,
                                              const float* __restrict__ W,
                                              const float* __restrict__ b,
                                              float* __restrict__ out, int n) {
    int wave = (blockIdx.x * blockDim.x + threadIdx.x) >> 5;
    int lane = threadIdx.x & 31;
    if (wave >= n) return;
    float4 hv = ((const float4*)(h + (size_t)wave * EMBED))[lane];
    float4 wv = ((const float4*)W)[lane];
    float s = hv.x * wv.x + hv.y * wv.y + hv.z * wv.z + hv.w * wv.w;
    #pragma unroll
    for (int off = 16; off > 0; off >>= 1) s += __shfl_down(s, off, 32);
    if (lane == 0) out[wave] = s + b[0];
}

// ---------------- orchestration ----------------
extern "C" void kernel_launch(void* const* d_in, const int* in_sizes, int n_in,
                              void* d_out, int out_size, void* d_ws, size_t ws_size,
                              hipStream_t stream) {
    const float* nf    = (const float*)d_in[0];
    const int*   src   = (const int*)d_in[1];
    const int*   dst   = (const int*)d_in[2];
    const float* W_emb = (const float*)d_in[3];
    const float* b_emb = (const float*)d_in[4];
    const float* W_g   = (const float*)d_in[5];   // [3,128,128]
    const float* b_g   = (const float*)d_in[6];   // [3,128]
    const float* W_o1  = (const float*)d_in[7];
    const float* b_o1  = (const float*)d_in[8];
    const float* W_o2  = (const float*)d_in[9];   // [128,1]
    const float* b_o2  = (const float*)d_in[10];
    float* out = (float*)d_out;
    (void)in_sizes; (void)n_in; (void)out_size; (void)ws_size;

    float* ws = (float*)d_ws;
    const size_t NF = (size_t)N_NODES * EMBED;   // 6.4M floats
    float* norm_src = ws;                         // [N]  (deg -> norm in place)
    float* norm_dst = ws + N_NODES;               // [N]
    float* bufA     = ws + 2 * N_NODES;           // [N,128]
    float* bufB     = bufA + NF;                  // [N,128]
    float* agg      = bufB + NF;                  // [N,128]

    const int T = 256;
    // degrees + norms (recomputed each call: deterministic, no cached state)
    k_zero<<<(2 * N_NODES + T - 1) / T, T, 0, stream>>>(norm_src, 2 * N_NODES);
    k_degrees<<<(N_EDGES + T - 1) / T, T, 0, stream>>>(src, dst, norm_src, norm_dst, N_EDGES);
    k_norm<<<(N_NODES + T - 1) / T, T, 0, stream>>>(norm_src, norm_dst, N_NODES);

    // embed GEMM -> bufA
    k_embed<<<(ROW_TILES * 32 + T - 1) / T, T, 0, stream>>>(nf, W_emb, b_emb, bufA);

    float* hin = bufA;
    float* hout = bufB;
    for (int l = 0; l < 3; ++l) {
        k_zero<<<((int)NF + T - 1) / T, T, 0, stream>>>(agg, (int)NF);
        k_scatter<<<(N_EDGES * 32 + T - 1) / T, T, 0, stream>>>(hin, norm_src, src, dst,
                                                                agg, N_EDGES);
        k_gemm128<<<1024, T, 0, stream>>>(agg, norm_dst, W_g + (size_t)l * EMBED * EMBED,
                                          b_g + (size_t)l * EMBED, hout, ROW_TILES, 1);
        float* tmp = hin; hin = hout; hout = tmp;
    }

    // h = relu(h @ W_o1 + b_o1)
    k_gemm128<<<1024, T, 0, stream>>>(hin, (const float*)nullptr, W_o1, b_o1, hout,
                                      ROW_TILES, 1);
    // out = h @ W_o2 + b_o2
    k_out2<<<(N_NODES * 32 + T - 1) / T, T, 0, stream>>>(hout, W_o2, b_o2, out, N_NODES);
}